// ShapeletsCNN_18287970747087
// MI455X (gfx1250) — compile-verified
//
#include <hip/hip_runtime.h>
#include <math.h>

typedef float v2f __attribute__((ext_vector_type(2)));
typedef float v8f __attribute__((ext_vector_type(8)));

#define S_SH   128
#define D_CH   3
#define L_SH   16
#define K_TOT  48            // D*L
#define T_LEN  16384
#define T_OUT  (T_LEN - L_SH + 1)   // 16369
#define B_N    32
#define H_DIM  512
#define C_DIM  10
#define TT     512           // t positions per block
#define NCHUNK ((T_OUT + TT - 1) / TT)  // 32
#define XS_LEN (TT + L_SH)   // 528 staged x samples per channel

__device__ __forceinline__ void atomicMinFloat(float* addr, float val) {
    unsigned int* ua = reinterpret_cast<unsigned int*>(addr);
    unsigned int cur = __float_as_uint(*addr);
    while (__uint_as_float(cur) > val) {
        unsigned int prev = atomicCAS(ua, cur, __float_as_uint(val));
        if (prev == cur) break;
        cur = prev;
    }
}

__global__ void init_inf_kernel(float* __restrict__ p, int n) {
    int i = blockIdx.x * blockDim.x + threadIdx.x;
    if (i < n) p[i] = INFINITY;
}

// One block: (t-chunk, batch). 8 waves; wave w owns shapelets [16w, 16w+16).
// dist[b,s] = min_t ( wsq[b,t] + ssq[s] - 2 * cross_corr[b,s,t] )
// The whole affine epilogue is fused into the WMMA chain:
//   B fragments hold -2*shapelet, and a 13th WMMA with
//   A cols {wsq[m], 1, 0, 0} x B rows {1, ssq[n], 0, 0}
//   adds wsq + ssq, so c[r] is the distance directly.
// Invalid t rows carry wsq = +INF, which propagates to +INF distances
// (no 0*INF anywhere: INF entries only multiply the constant 1).
__global__ __launch_bounds__(256) void shapelet_min_kernel(
    const float* __restrict__ x,      // (B, D, T)
    const float* __restrict__ shp,    // (S, 1, D, L) -> contiguous K=48 per s
    float* __restrict__ dist_out)     // (B, S), pre-set to +INF
{
    __shared__ float xs[D_CH][XS_LEN];
    __shared__ float wsq[TT];

    const int chunk = blockIdx.x;
    const int b     = blockIdx.y;
    const int t0    = chunk * TT;
    const int tid   = threadIdx.x;

    // Stage x window for this chunk (zero-pad past end of series).
    for (int i = tid; i < D_CH * XS_LEN; i += 256) {
        int d = i / XS_LEN, o = i - d * XS_LEN;
        int t = t0 + o;
        xs[d][o] = (t < T_LEN)
            ? x[(size_t)b * (D_CH * T_LEN) + (size_t)d * T_LEN + t]
            : 0.0f;
    }
    __syncthreads();

    // Window energy per output position; +INF poison for t >= T_OUT makes
    // the main loop fully branchless.
    for (int i = tid; i < TT; i += 256) {
        float acc = 0.0f;
        #pragma unroll
        for (int d = 0; d < D_CH; ++d) {
            #pragma unroll
            for (int l = 0; l < L_SH; ++l) {
                float v = xs[d][i + l];
                acc += v * v;
            }
        }
        wsq[i] = ((t0 + i) < T_OUT) ? acc : INFINITY;
    }
    __syncthreads();

    const int lane = tid & 31;
    const int n    = lane & 15;   // N index; also M index for A fragments
    const int h    = lane >> 4;   // lane half
    const int s0   = (tid >> 5) * 16;

    // B fragments (scaled by -2): register-resident for the whole chunk.
    // Fragment j covers K rows [4j, 4j+4): VGPR0 <-> K=4j+2h, VGPR1 <-> K=4j+2h+1.
    v2f bf[12];
    float ssq_p = 0.0f;
    #pragma unroll
    for (int j = 0; j < 12; ++j) {
        int kb = 4 * j + 2 * h;
        const float* p = shp + (size_t)(s0 + n) * K_TOT + kb;
        float p0 = p[0], p1 = p[1];
        ssq_p += p0 * p0 + p1 * p1;
        bf[j].x = -2.0f * p0;
        bf[j].y = -2.0f * p1;
    }
    // Each lane holds half the K values of shapelet (s0+n); other half in lane^16.
    const float ssq = ssq_p + __shfl_xor(ssq_p, 16, 32);

    // Constant B fragment of the fused epilogue WMMA: rows {1, ssq, 0, 0}.
    v2f bf_ep;
    bf_ep.x = h ? 0.0f : 1.0f;
    bf_ep.y = h ? 0.0f : ssq;

    float runmin = INFINITY;

    // 32 M-tiles of 16 output positions each.
    for (int tile = 0; tile < TT / 16; ++tile) {
        const int tb = tile * 16;

        v8f c = {0.f, 0.f, 0.f, 0.f, 0.f, 0.f, 0.f, 0.f};
        #pragma unroll
        for (int j = 0; j < 12; ++j) {
            int kb = 4 * j + 2 * h;         // even -> both elems in same channel row
            int d  = kb >> 4;
            int l  = kb & 15;
            v2f a;
            a.x = xs[d][tb + n + l];        // A[m=n, K=kb]
            a.y = xs[d][tb + n + l + 1];    // A[m=n, K=kb+1]
            c = __builtin_amdgcn_wmma_f32_16x16x4_f32(
                    false, a, false, bf[j], (short)0, c, false, false);
        }

        // Fused epilogue WMMA: adds wsq[m] + ssq[n] to every element.
        float wv = wsq[tb + n];             // row m = n for A fragments
        v2f a_ep;
        a_ep.x = h ? 0.0f : wv;
        a_ep.y = h ? 0.0f : 1.0f;
        c = __builtin_amdgcn_wmma_f32_16x16x4_f32(
                false, a_ep, false, bf_ep, (short)0, c, false, false);

        // c[r] is now the distance for (m = r + 8h, n); pure min reduction.
        #pragma unroll
        for (int r = 0; r < 8; ++r) {
            runmin = fminf(runmin, c[r]);
        }
    }

    // Combine the two lane halves (same N, different M sets), then commit.
    runmin = fminf(runmin, __shfl_xor(runmin, 16, 32));
    if (h == 0) atomicMinFloat(&dist_out[b * S_SH + s0 + n], runmin);
}

// Per-batch tiny MLP: h = relu(dist @ W1^T + b1); cls = h @ W2^T + b2.
__global__ __launch_bounds__(512) void mlp_kernel(
    const float* __restrict__ dist,   // (B, S)
    const float* __restrict__ W1,     // (H, S)
    const float* __restrict__ b1,     // (H)
    const float* __restrict__ W2,     // (C, H)
    const float* __restrict__ b2,     // (C)
    float* __restrict__ cls_out)      // (B, C)
{
    __shared__ float ds[S_SH];
    __shared__ float hs[H_DIM];
    const int b = blockIdx.x, tid = threadIdx.x;

    if (tid < S_SH) ds[tid] = dist[b * S_SH + tid];
    __syncthreads();

    float acc = b1[tid];
    const float* w = W1 + (size_t)tid * S_SH;
    #pragma unroll 8
    for (int k = 0; k < S_SH; ++k) acc += ds[k] * w[k];
    hs[tid] = fmaxf(acc, 0.0f);
    __syncthreads();

    if (tid < C_DIM) {
        float a2 = b2[tid];
        const float* w2 = W2 + (size_t)tid * H_DIM;
        #pragma unroll 8
        for (int k = 0; k < H_DIM; ++k) a2 += hs[k] * w2[k];
        cls_out[b * C_DIM + tid] = a2;
    }
}

extern "C" void kernel_launch(void* const* d_in, const int* in_sizes, int n_in,
                              void* d_out, int out_size, void* d_ws, size_t ws_size,
                              hipStream_t stream) {
    const float* x   = (const float*)d_in[0];
    const float* shp = (const float*)d_in[1];
    const float* W1  = (const float*)d_in[2];
    const float* b1  = (const float*)d_in[3];
    const float* W2  = (const float*)d_in[4];
    const float* b2  = (const float*)d_in[5];

    float* dist = (float*)d_out;                 // first B*S floats of output
    float* cls  = dist + B_N * S_SH;             // then B*C floats

    // Must re-init every call (harness does not re-poison between replays).
    init_inf_kernel<<<(B_N * S_SH + 255) / 256, 256, 0, stream>>>(dist, B_N * S_SH);

    dim3 grid(NCHUNK, B_N);
    shapelet_min_kernel<<<grid, 256, 0, stream>>>(x, shp, dist);

    mlp_kernel<<<B_N, H_DIM, 0, stream>>>(dist, W1, b1, W2, b2, cls);
}